// GDLoss_71786083386087
// MI455X (gfx1250) — compile-verified
//
#include <hip/hip_runtime.h>

typedef __attribute__((ext_vector_type(2))) float v2f;
typedef __attribute__((ext_vector_type(8))) float v8f;

#define DEG2RAD 0.017453292519943295f
#define EARTH_R 6371.009f

// ---------------------------------------------------------------------------
// Wave-level sum using the CDNA5 matrix pipe.
// A (16x4 f32) holds the 32 per-lane partials in VGPR0 (lanes 0-15 -> K=0,
// lanes 16-31 -> K=2), VGPR1 = 0.  B = ones(4x16).  D[m][n] = acc[m]+acc[m+16]
// for every n, so summing the 8 D VGPRs gives sum(rows 0-7) on lanes 0-15 and
// sum(rows 8-15) on lanes 16-31; xor-shuffle across the half-wave completes it.
// Requires EXEC == all ones (call only from non-divergent full-wave code).
// ---------------------------------------------------------------------------
__device__ __forceinline__ float wave_sum_wmma(float x) {
    v2f a; a.x = x;    a.y = 0.0f;
    v2f b; b.x = 1.0f; b.y = 1.0f;
    v8f c = {};
    v8f d = __builtin_amdgcn_wmma_f32_16x16x4_f32(
        /*neg_a=*/false, a, /*neg_b=*/false, b,
        /*c_mod=*/(short)0, c, /*reuse_a=*/false, /*reuse_b=*/false);
    float s = d[0] + d[1] + d[2] + d[3] + d[4] + d[5] + d[6] + d[7];
    s += __shfl_xor(s, 16, 32);   // combine the two half-wave row groups
    return s;                     // full 32-lane sum, valid on every lane
}

// Block reduction: 8 waves (256 threads) -> one float.
__device__ __forceinline__ float block_sum(float x, float* lds8) {
    const int lane = threadIdx.x & 31;
    const int wave = threadIdx.x >> 5;
    float w = wave_sum_wmma(x);
    if (lane == 0) lds8[wave] = w;
    __syncthreads();
    float t = 0.0f;
    if (threadIdx.x == 0) {
#pragma unroll
        for (int i = 0; i < 8; ++i) t += lds8[i];
    }
    return t;  // valid on thread 0 only
}

// ---------------------------------------------------------------------------
// Per-element great-circle distance (Vincenty sphere special case).
// ---------------------------------------------------------------------------
__device__ __forceinline__ float gd_elem(
    float plat_, float plng_, float tlat_, float tlng_,
    float slat,  float slng,
    float ms0, float ms1, float ss0, float ss1,
    float m0,  float m1,  float s0,  float s1)
{
    const float site_lat = fmaf(slat, s0, m0);
    const float site_lng = fmaf(slng, s1, m1);
    const float plat = fmaf(plat_, ss0, ms0) + site_lat;
    const float plng = fmaf(plng_, ss1, ms1) + site_lng;
    const float tlat = fmaf(tlat_, ss0, ms0) + site_lat;
    const float tlng = fmaf(tlng_, ss1, ms1) + site_lng;

    const float lat1 = fminf(fmaxf(plat, -90.0f), 90.0f) * DEG2RAD;
    const float lng1 = plng * DEG2RAD;
    const float lat2 = tlat * DEG2RAD;
    const float lng2 = tlng * DEG2RAD;

    float sl1, cl1, sl2, cl2, sd, cd;
    __sincosf(lat1, &sl1, &cl1);           // v_sin_f32 / v_cos_f32 (args < ~2 rad)
    __sincosf(lat2, &sl2, &cl2);
    __sincosf(lng2 - lng1, &sd, &cd);

    const float t1  = cl2 * sd;
    const float t2  = fmaf(cl1, sl2, -sl1 * cl2 * cd);
    const float num = sqrtf(fmaf(t1, t1, t2 * t2));
    const float den = fmaf(sl1, sl2, cl1 * cl2 * cd);
    return EARTH_R * atan2f(num, den);
}

// ---------------------------------------------------------------------------
// Pass 1: grid-stride over float4 (two [lat,lng] points per lane per array),
// fully coalesced global_load_b128 streams; block partial -> d_ws[blockIdx].
// ---------------------------------------------------------------------------
__global__ void __launch_bounds__(256)
gd_main_kernel(const float* __restrict__ pred, const float* __restrict__ tru,
               const float* __restrict__ mean_speed, const float* __restrict__ std_speed,
               const float* __restrict__ site, const float* __restrict__ mean,
               const float* __restrict__ stdv,
               float* __restrict__ partials, long long B)
{
    __shared__ float lds8[8];

    const float ms0 = mean_speed[0], ms1 = mean_speed[1];
    const float ss0 = std_speed[0],  ss1 = std_speed[1];
    const float m0  = mean[0],       m1  = mean[1];
    const float s0  = stdv[0],       s1  = stdv[1];

    const long long npair  = B >> 1;
    const float4* __restrict__ p4 = (const float4*)pred;
    const float4* __restrict__ t4 = (const float4*)tru;
    const float4* __restrict__ g4 = (const float4*)site;

    const long long stride = (long long)gridDim.x * blockDim.x;
    float acc = 0.0f;

    for (long long i = (long long)blockIdx.x * blockDim.x + threadIdx.x;
         i < npair; i += stride) {
        const float4 p = p4[i];
        const float4 t = t4[i];
        const float4 g = g4[i];
        acc += gd_elem(p.x, p.y, t.x, t.y, g.x, g.y,
                       ms0, ms1, ss0, ss1, m0, m1, s0, s1);
        acc += gd_elem(p.z, p.w, t.z, t.w, g.z, g.w,
                       ms0, ms1, ss0, ss1, m0, m1, s0, s1);
    }

    // Odd-B tail (not hit for B = 8388608, kept for generality).
    if ((B & 1) && blockIdx.x == 0 && threadIdx.x == 0) {
        const long long j = (B - 1) * 2;
        acc += gd_elem(pred[j], pred[j + 1], tru[j], tru[j + 1],
                       site[j], site[j + 1],
                       ms0, ms1, ss0, ss1, m0, m1, s0, s1);
    }

    const float bsum = block_sum(acc, lds8);
    if (threadIdx.x == 0) partials[blockIdx.x] = bsum;
}

// ---------------------------------------------------------------------------
// Pass 2: single block reduces the partials deterministically, scales by 1/B.
// ---------------------------------------------------------------------------
__global__ void __launch_bounds__(256)
gd_finalize_kernel(const float* __restrict__ partials, int n,
                   float* __restrict__ out, float inv_b)
{
    __shared__ float lds8[8];
    float acc = 0.0f;
    for (int i = threadIdx.x; i < n; i += 256) acc += partials[i];
    const float total = block_sum(acc, lds8);
    if (threadIdx.x == 0) out[0] = total * inv_b;
}

// ---------------------------------------------------------------------------
extern "C" void kernel_launch(void* const* d_in, const int* in_sizes, int n_in,
                              void* d_out, int out_size, void* d_ws, size_t ws_size,
                              hipStream_t stream)
{
    (void)n_in; (void)out_size;
    const float* pred       = (const float*)d_in[0];
    const float* tru        = (const float*)d_in[1];
    const float* mean_speed = (const float*)d_in[2];
    const float* std_speed  = (const float*)d_in[3];
    const float* site       = (const float*)d_in[4];
    const float* mean       = (const float*)d_in[5];
    const float* stdv       = (const float*)d_in[6];
    float* out = (float*)d_out;
    float* partials = (float*)d_ws;

    const long long B = (long long)in_sizes[0] / 2;   // pred_ is [B,2]

    int nblk = 2048;                                   // 524k threads: saturates HBM streams
    const long long maxblk = (long long)(ws_size / sizeof(float));
    if (maxblk > 0 && nblk > maxblk) nblk = (int)maxblk;
    if (nblk < 1) nblk = 1;

    gd_main_kernel<<<nblk, 256, 0, stream>>>(pred, tru, mean_speed, std_speed,
                                             site, mean, stdv, partials, B);
    gd_finalize_kernel<<<1, 256, 0, stream>>>(partials, nblk, out,
                                              1.0f / (float)B);
}